// CumulativeGroupNorm2d_67336497267012
// MI455X (gfx1250) — compile-verified
//
#include <hip/hip_runtime.h>
#include <cstdint>

#define EPS 1e-5f

// Problem geometry (fixed by the reference): x is [B, C, T, F] f32.
constexpr int T = 1024;
constexpr int F = 128;
constexpr int CT = 16;                    // timesteps per LDS tile
constexpr int CHUNK_F4 = CT * F / 4;      // 512 float4 per tile (8 KB)
constexpr int NBUF = 3;                   // triple buffer -> depth-3 pipeline
constexpr int NCHUNK = T / CT;            // 64 tiles per row
constexpr int ISSUES = CHUNK_F4 / 32;     // 16 async b128 issues per tile per wave

typedef __attribute__((ext_vector_type(4))) float v4f;

// gfx1250 async global->LDS copy: 16 B/lane, 512 B per wave-issue, ASYNCcnt
// tracked. Streaming data (zero reuse) -> non-temporal hint.
__device__ __forceinline__ void async_ld_b128(unsigned lds_off, const void* gptr) {
  asm volatile("global_load_async_to_lds_b128 %0, %1, off th:TH_LOAD_NT"
               :
               : "v"(lds_off), "v"((unsigned long long)(uintptr_t)gptr)
               : "memory");
}

__global__ __launch_bounds__(32)
void cgn_scan_kernel(const float* __restrict__ x,
                     const float* __restrict__ wgt,
                     const float* __restrict__ bsc,
                     float* __restrict__ out, int C) {
  __shared__ float lds[NBUF * CT * F];   // 24 KB

  const int lane = threadIdx.x;          // lane owns f = lane*4 .. lane*4+3
  const int row  = blockIdx.x;           // flattened (b*C + c)
  const int c    = row & (C - 1);        // C = 256 (power of two)

  const float w = wgt[c];
  const float b = bsc[c];

  const v4f* __restrict__ xin  = (const v4f*)x + (size_t)row * (size_t)(T * F / 4);
  v4f* __restrict__       xout = (v4f*)out     + (size_t)row * (size_t)(T * F / 4);

  // Wave-relative LDS byte offset = low 32 bits of generic shared address.
  const unsigned lds_base = (unsigned)(uintptr_t)(&lds[0]);
  const v4f* lds4 = (const v4f*)(&lds[0]);

  float s[4] = {0.f, 0.f, 0.f, 0.f};
  float q[4] = {0.f, 0.f, 0.f, 0.f};
  float n = 0.f;                          // running count as exact f32 (<=1024)

  auto issue = [&](int ci, int bi) {
    const v4f* gp = xin + ci * CHUNK_F4 + lane;
    const unsigned lo = lds_base + (unsigned)(bi * CHUNK_F4 * 16) + (unsigned)(lane * 16);
#pragma unroll
    for (int k = 0; k < ISSUES; ++k)
      async_ld_b128(lo + (unsigned)(k * 512), gp + k * 32);
  };

  auto compute = [&](int ci, int bi) {
#pragma unroll
    for (int t = 0; t < CT; ++t) {
      v4f v = lds4[bi * CHUNK_F4 + t * (F / 4) + lane];   // ds_load_b128
      const int tt = ci * CT + t;
      n += 1.0f;
      const float en2 = (n * n) * EPS;                    // eps * n^2, per-t uniform
      v4f o;
#pragma unroll
      for (int j = 0; j < 4; ++j) {
        // division-free scan-normalize:
        // (x-mean)*rsqrt(var+eps) == (n*x - s) * rsqrt(n*q - s^2 + eps*n^2)
        s[j] += v[j];
        q[j] = fmaf(v[j], v[j], q[j]);
        const float a  = fmaf(n, v[j], -s[j]);
        const float d  = fmaf(n, q[j], -(s[j] * s[j]));
        const float r  = rsqrtf(d + en2);
        o[j] = fmaf(a * r, w, b);
      }
      __builtin_nontemporal_store(o, &xout[tt * (F / 4) + lane]);  // b128 NT store
    }
  };

  // Depth-3 pipeline: keep ~3 tiles (48 x b128) of async loads in flight per
  // wave so 1024 waves cover the HBM bandwidth-delay product (~18-24 MB).
  issue(0, 0);
  issue(1, 1);
  int ib = 2;   // buffer receiving the next issued tile
  int cb = 0;   // buffer holding the tile to compute
#pragma unroll 1
  for (int ci = 0; ci < NCHUNK - 2; ++ci) {
    issue(ci + 2, ib);
    ib = (ib == NBUF - 1) ? 0 : ib + 1;
    // outstanding = tiles ci, ci+1, ci+2 (48); <=32 left => tile ci resident.
    asm volatile("s_wait_asynccnt 32" ::: "memory");
    compute(ci, cb);
    cb = (cb == NBUF - 1) ? 0 : cb + 1;
  }
  asm volatile("s_wait_asynccnt 16" ::: "memory");
  compute(NCHUNK - 2, cb);
  cb = (cb == NBUF - 1) ? 0 : cb + 1;
  asm volatile("s_wait_asynccnt 0" ::: "memory");
  compute(NCHUNK - 1, cb);
}

extern "C" void kernel_launch(void* const* d_in, const int* in_sizes, int n_in,
                              void* d_out, int out_size, void* d_ws, size_t ws_size,
                              hipStream_t stream) {
  const float* x   = (const float*)d_in[0];
  const float* wgt = (const float*)d_in[1];   // [1,C,1,1] -> C floats
  const float* bsc = (const float*)d_in[2];   // [1,C,1,1] -> C floats
  float* out = (float*)d_out;

  const int C    = in_sizes[1];                                        // 256
  const int rows = (int)((long long)in_sizes[0] / (long long)(T * F)); // B*C = 1024

  cgn_scan_kernel<<<rows, 32, 0, stream>>>(x, wgt, bsc, out, C);
}